// Drug_cell_encoder_10050223473328
// MI455X (gfx1250) — compile-verified
//
#include <hip/hip_runtime.h>
#include <hip/hip_bf16.h>

typedef __attribute__((ext_vector_type(2))) float v2f;
typedef __attribute__((ext_vector_type(8))) float v8f;

#define DIMD 256
#define TM 64
#define TN 64
#define TK 64
#define SP 160   // pair-row stride (floats): 160 % 64 == 32 -> half-waves hit disjoint bank halves

__device__ __forceinline__ void atomAddF32(float* p, float v) {
    __hip_atomic_fetch_add(p, v, __ATOMIC_RELAXED, __HIP_MEMORY_SCOPE_AGENT);
}

// ---- degree: one thread per edge, atomic count into deg[dst] ----
__global__ void deg_kernel(const int* __restrict__ dst, float* __restrict__ deg, int E) {
    int e = blockIdx.x * blockDim.x + threadIdx.x;
    if (e < E) atomAddF32(&deg[dst[e]], 1.0f);
}

// ---- inv_deg[i] = 1 / max(deg[i], 1) ----
__global__ void inv_kernel(const float* __restrict__ deg, float* __restrict__ inv, int N) {
    int i = blockIdx.x * blockDim.x + threadIdx.x;
    if (i < N) inv[i] = 1.0f / fmaxf(deg[i], 1.0f);
}

// ---- scatter: agg[dst] += cur[src]; 64 threads per edge, float4 chunks ----
__global__ void scatter_kernel(const float* __restrict__ cur,
                               const int* __restrict__ src,
                               const int* __restrict__ dst,
                               float* __restrict__ agg, int E) {
    long long gid = (long long)blockIdx.x * blockDim.x + threadIdx.x;
    int e = (int)(gid >> 6);
    if (e >= E) return;
    int c = ((int)gid & 63) << 2;                 // 0..252, step 4 floats
    int s = src[e], d = dst[e];
    const float4 v = *(const float4*)(cur + (size_t)s * DIMD + c);
    float* p = agg + (size_t)d * DIMD + c;
    atomAddF32(p + 0, v.x);
    atomAddF32(p + 1, v.y);
    atomAddF32(p + 2, v.z);
    atomAddF32(p + 3, v.w);
}

// ---- fused: h = 0.1*x0 + 0.9*inv_deg*agg;  out = relu(h @ W)  via WMMA f32 ----
// Block: 256 threads (8 wave32s) -> 64x64 output tile; each wave owns 16x32.
// LDS layout: element (k, col) lives at [(k>>1)*SP + col*2 + (k&1)]  so each
// lane's WMMA fragment {K, K+1} is one aligned ds_load_b64, no register repack.
__global__ __launch_bounds__(256) void gcn_gemm_kernel(
        const float* __restrict__ x0, const float* __restrict__ agg,
        const float* __restrict__ invdeg, const float* __restrict__ W,
        float* __restrict__ out, int nrows) {
    __shared__ __align__(16) float As[(TK / 2) * SP];  // h tile, pair-interleaved [kpair][M*2+par]
    __shared__ __align__(16) float Bs[(TK / 2) * SP];  // W chunk, pair-interleaved [kpair][n*2+par]

    const int t    = threadIdx.x;
    const int lane = t & 31;
    const int w    = t >> 5;
    const int rb   = (w >> 1) * 16;          // wave row offset in tile (0/16/32/48)
    const int nb   = (w & 1) * 32;           // wave col offset in tile (0/32)
    const int rowBlock = blockIdx.x * TM;
    const int colBlock = blockIdx.y * TN;

    v8f c0 = {};
    v8f c1 = {};

    // A staging assignment: thread -> (tile row alr, 16-wide k segment alc)
    const int alr = t >> 2;                  // 0..63  (M row)
    const int alc = (t & 3) * 16;            // 0/16/32/48 (k offset, even)
    const int grow = rowBlock + alr;
    const bool rvalid = grow < nrows;
    const float inv = rvalid ? invdeg[grow] : 0.0f;

    // B staging assignment: thread -> (k-pair bp, 8-wide col segment bcs)
    const int bp  = t >> 3;                  // 0..31  (k pair)
    const int bcs = (t & 7) * 8;             // 0..56  (n offset)

    // per-lane fragment offsets (pair row += lane>>4 gives K+2/K+3 half)
    const int m    = lane & 15;
    const int prow = lane >> 4;              // 0 or 1
    const int aoff = prow * SP + (rb + m) * 2;
    const int boff = prow * SP + (nb + m) * 2;

    for (int kb = 0; kb < DIMD; kb += TK) {
        // ---- stage A = h = 0.1*x0 + 0.9*inv_deg*agg (fused, never hits HBM) ----
        #pragma unroll
        for (int j = 0; j < 16; j += 4) {
            float4 xa = {0.f, 0.f, 0.f, 0.f};
            float4 ga = {0.f, 0.f, 0.f, 0.f};
            if (rvalid) {
                xa = *(const float4*)(x0  + (size_t)grow * DIMD + kb + alc + j);
                ga = *(const float4*)(agg + (size_t)grow * DIMD + kb + alc + j);
            }
            const int k0 = alc + j;          // even
            float2 h01, h23;
            h01.x = 0.1f * xa.x + 0.9f * inv * ga.x;
            h01.y = 0.1f * xa.y + 0.9f * inv * ga.y;
            h23.x = 0.1f * xa.z + 0.9f * inv * ga.z;
            h23.y = 0.1f * xa.w + 0.9f * inv * ga.w;
            *(float2*)&As[(k0 >> 1) * SP + alr * 2]       = h01;
            *(float2*)&As[((k0 >> 1) + 1) * SP + alr * 2] = h23;
        }
        // ---- stage B: interleave rows k=2bp and k=2bp+1 of W into pair layout ----
        #pragma unroll
        for (int j = 0; j < 8; j += 4) {
            const float4 w0 = *(const float4*)(W + (size_t)(kb + 2 * bp)     * DIMD + colBlock + bcs + j);
            const float4 w1 = *(const float4*)(W + (size_t)(kb + 2 * bp + 1) * DIMD + colBlock + bcs + j);
            float* bpw = &Bs[bp * SP + (bcs + j) * 2];
            *(float2*)(bpw + 0) = make_float2(w0.x, w1.x);
            *(float2*)(bpw + 2) = make_float2(w0.y, w1.y);
            *(float2*)(bpw + 4) = make_float2(w0.z, w1.z);
            *(float2*)(bpw + 6) = make_float2(w0.w, w1.w);
        }
        __syncthreads();

        // ---- 16 K-steps: 1 A + 2 B ds_load_b64, 2x V_WMMA_F32_16X16X4_F32 ----
        #pragma unroll
        for (int ks = 0; ks < TK; ks += 4) {
            const int pr = (ks >> 1) * SP;   // base pair-row offset for this step
            v2f a  = *(const v2f*)&As[pr + aoff];
            v2f b0 = *(const v2f*)&Bs[pr + boff];
            v2f b1 = *(const v2f*)&Bs[pr + boff + 32];
            c0 = __builtin_amdgcn_wmma_f32_16x16x4_f32(false, a, false, b0,
                                                       (short)0, c0, false, false);
            c1 = __builtin_amdgcn_wmma_f32_16x16x4_f32(false, a, false, b1,
                                                       (short)0, c1, false, false);
        }
        __syncthreads();
    }

    // ---- epilogue: ReLU + store (C layout: VGPR r -> M=r / M=r+8, N=lane%16) ----
    const int mh = prow * 8;
    #pragma unroll
    for (int r = 0; r < 8; ++r) {
        int row = rowBlock + rb + mh + r;
        if (row < nrows) {
            size_t base = (size_t)row * DIMD + colBlock + nb;
            out[base + m]      = fmaxf(c0[r], 0.0f);
            out[base + 16 + m] = fmaxf(c1[r], 0.0f);
        }
    }
}

extern "C" void kernel_launch(void* const* d_in, const int* in_sizes, int n_in,
                              void* d_out, int out_size, void* d_ws, size_t ws_size,
                              hipStream_t stream) {
    const float* x0    = (const float*)d_in[0];
    const int*   edges = (const int*)d_in[1];      // [2, E]: src row then dst row
    const float* W     = (const float*)d_in[2];
    float*       out   = (float*)d_out;

    const int N = in_sizes[0] / DIMD;
    const int E = in_sizes[1] / 2;
    const int* src = edges;
    const int* dst = edges + E;

    // workspace layout
    float* agg = (float*)d_ws;                     // N*256
    float* cur = agg + (size_t)N * DIMD;           // N*256
    float* deg = cur + (size_t)N * DIMD;           // N
    float* inv = deg + N;                          // N

    // degree normalization: edge-structure only -> compute once per call
    hipMemsetAsync(deg, 0, (size_t)N * sizeof(float), stream);
    deg_kernel<<<(E + 255) / 256, 256, 0, stream>>>(dst, deg, E);
    inv_kernel<<<(N + 255) / 256, 256, 0, stream>>>(deg, inv, N);

    const long long sthreads = (long long)E * 64;
    const int sblocks = (int)((sthreads + 255) / 256);
    dim3 ggrid((N + TM - 1) / TM, DIMD / TN);

    const float* in = x0;
    for (int layer = 0; layer < 5; ++layer) {
        hipMemsetAsync(agg, 0, (size_t)N * DIMD * sizeof(float), stream);
        scatter_kernel<<<sblocks, 256, 0, stream>>>(in, src, dst, agg, E);
        float* dest = (layer == 4) ? out : cur;    // GEMM reads only agg/x0 -> cur in-place OK
        gcn_gemm_kernel<<<ggrid, 256, 0, stream>>>(x0, agg, inv, W, dest, N);
        in = cur;
    }
}